// MultiBoxLoss_83537113907246
// MI455X (gfx1250) — compile-verified
//
#include <hip/hip_runtime.h>
#include <hip/hip_bf16.h>
#include <stdint.h>

// MultiBoxLoss (SSD) for MI455X / gfx1250.
// B=64, P=24564, G=32, C=81. Bandwidth-dominated by one 509MB pass over
// conf_data; per-row sum(exp) reduction done with V_WMMA_F32_16X16X4_F32
// (full f32 precision, B = ones -> row sums). conf_data staged to LDS with
// GLOBAL_LOAD_ASYNC_TO_LDS_B128 (ASYNCcnt) when the builtin is available.
// Hard-negative mining via deterministic 4-pass radix select (top-K sum is
// tie-invariant, so it matches the reference's double-argsort selection).

#define BB 64
#define PP 24564
#define GG 32
#define CC 81
#define NEG_POS_K 3

#define PBLK ((PP + 255) / 256)   // 96 blocks of 256 over priors
#define CE_BLOCKS 1024
#define ROWS (BB * PP)            // 1,572,096
#define TILE 128                  // rows per block-iteration (8 waves * 16)
#define NTILES (ROWS / TILE)      // 12282 exactly

// ---- gfx1250 async global->LDS staging (guarded; falls back cleanly) -----
#if defined(__HIP_DEVICE_COMPILE__)
#  if __has_builtin(__builtin_amdgcn_global_load_async_to_lds_b128)
#    define HAVE_ASYNC_LDS 1
#  endif
#endif
#ifndef HAVE_ASYNC_LDS
#  define HAVE_ASYNC_LDS 0
#endif

#if HAVE_ASYNC_LDS
// Builtin expects pointers to a 16-byte int vector in AS1 (global) / AS3 (LDS)
// per the clang diagnostic: 'vector_size(4*sizeof(int)) int __device__ *'.
typedef int i32x4 __attribute__((vector_size(4 * sizeof(int))));
typedef __attribute__((address_space(1))) i32x4 gi32x4;  // global
typedef __attribute__((address_space(3))) i32x4 li32x4;  // LDS
static __device__ __forceinline__ void wait_asynccnt0() {
#  if __has_builtin(__builtin_amdgcn_s_wait_asynccnt)
  __builtin_amdgcn_s_wait_asynccnt(0);
#  else
  asm volatile("s_wait_asynccnt 0x0" ::: "memory");
#  endif
}
#endif

typedef float v2f __attribute__((ext_vector_type(2)));
typedef float v8f __attribute__((ext_vector_type(8)));

static __device__ __forceinline__ float warp_sum_f(float v) {
#pragma unroll
  for (int off = 16; off; off >>= 1) v += __shfl_xor(v, off, 32);
  return v;
}

// ---------------------------------------------------------------- init ----
__global__ __launch_bounds__(256) void k_init(const float* __restrict__ dbox,
                                              float* __restrict__ pp4,
                                              unsigned long long* __restrict__ bp_key,
                                              int* __restrict__ num_pos) {
  int i = blockIdx.x * 256 + threadIdx.x;
  if (i < PP) {  // point_form(priors)
    float cx = dbox[i * 4 + 0], cy = dbox[i * 4 + 1];
    float w = dbox[i * 4 + 2], h = dbox[i * 4 + 3];
    pp4[i * 4 + 0] = cx - w * 0.5f;
    pp4[i * 4 + 1] = cy - h * 0.5f;
    pp4[i * 4 + 2] = cx + w * 0.5f;
    pp4[i * 4 + 3] = cy + h * 0.5f;
  }
  if (i < BB * GG) bp_key[i] = 0ULL;
  if (i < BB) num_pos[i] = 0;
}

// ------------------------------------------------------------- matching ---
// Per (b,p): best gt over 32 (first-max tiebreak == jnp.argmax axis=0).
// Per (b,g): best prior via wave-reduce + global atomicMax on packed
// (iou_bits<<32)|~p  (max of non-negative float bits is monotonic; ~p makes
// ties prefer the smallest prior index == jnp.argmax axis=1).
__global__ __launch_bounds__(256) void k_match(const float* __restrict__ targets,
                                               const float* __restrict__ pp4,
                                               float* __restrict__ bt_ov,
                                               int* __restrict__ bt_g,
                                               unsigned long long* __restrict__ bp_key) {
  __shared__ float st[GG][4];
  __shared__ float sarea[GG];
  int b = blockIdx.y, tid = threadIdx.x;
  if (tid < GG * 4) {
    int g = tid >> 2, k = tid & 3;
    st[g][k] = targets[(size_t)(b * GG + g) * 5 + k];
  }
  __syncthreads();
  if (tid < GG) sarea[tid] = (st[tid][2] - st[tid][0]) * (st[tid][3] - st[tid][1]);
  __syncthreads();

  int p = blockIdx.x * 256 + tid;
  bool act = p < PP;
  int pc = act ? p : (PP - 1);
  float px1 = pp4[pc * 4 + 0], py1 = pp4[pc * 4 + 1];
  float px2 = pp4[pc * 4 + 2], py2 = pp4[pc * 4 + 3];
  float areaB = (px2 - px1) * (py2 - py1);

  float best = -1.0f;
  int bestg = 0;
  for (int g = 0; g < GG; ++g) {
    float lx = fmaxf(st[g][0], px1), ly = fmaxf(st[g][1], py1);
    float rx = fminf(st[g][2], px2), ry = fminf(st[g][3], py2);
    float w = fmaxf(rx - lx, 0.f), h = fmaxf(ry - ly, 0.f);
    float inter = w * h;
    float iou = inter / (sarea[g] + areaB - inter);
    if (act && iou > best) { best = iou; bestg = g; }
    unsigned long long key =
        act ? ((((unsigned long long)__float_as_uint(iou)) << 32) |
               (unsigned)(~(unsigned)p))
            : 0ULL;
#pragma unroll
    for (int off = 16; off; off >>= 1) {
      unsigned long long o = __shfl_xor(key, off, 32);
      if (o > key) key = o;
    }
    if ((tid & 31) == 0) atomicMax(&bp_key[b * GG + g], key);
  }
  if (act) {
    bt_ov[(size_t)b * PP + p] = best;
    bt_g[(size_t)b * PP + p] = bestg;
  }
}

// Sequential per-batch force-match (last g wins on duplicate best priors).
__global__ void k_force(const unsigned long long* __restrict__ bp_key,
                        float* __restrict__ bt_ov, int* __restrict__ bt_g) {
  int b = blockIdx.x * blockDim.x + threadIdx.x;
  if (b >= BB) return;
  for (int g = 0; g < GG; ++g) {
    unsigned long long key = bp_key[b * GG + g];
    unsigned pr = ~(unsigned)(key & 0xFFFFFFFFull);
    bt_ov[(size_t)b * PP + pr] = 2.0f;
    bt_g[(size_t)b * PP + pr] = g;
  }
}

// -------------------------------------------- conf_t + localization loss --
__global__ __launch_bounds__(256) void k_conf_loc(const float* __restrict__ targets,
                                                  const float* __restrict__ dbox,
                                                  const float* __restrict__ loc_data,
                                                  const float* __restrict__ bt_ov,
                                                  const int* __restrict__ bt_g,
                                                  int* __restrict__ conf_t,
                                                  int* __restrict__ num_pos,
                                                  float* __restrict__ partial_l) {
  __shared__ float st[GG][4];
  __shared__ float slbl[GG];
  __shared__ float sred[8];
  int b = blockIdx.y, tid = threadIdx.x;
  if (tid < GG * 4) {
    int g = tid >> 2, k = tid & 3;
    st[g][k] = targets[(size_t)(b * GG + g) * 5 + k];
  }
  if (tid < GG) slbl[tid] = targets[(size_t)(b * GG + tid) * 5 + 4];
  __syncthreads();

  int p = blockIdx.x * 256 + tid;
  bool act = p < PP;
  float ll = 0.f;
  int cnt = 0;
  if (act) {
    size_t idx = (size_t)b * PP + p;
    float ov = bt_ov[idx];
    int g = bt_g[idx];
    int conf = (ov < 0.5f) ? 0 : ((int)slbl[g] + 1);
    conf_t[idx] = conf;
    if (conf > 0) {
      cnt = 1;
      float mx1 = st[g][0], my1 = st[g][1], mx2 = st[g][2], my2 = st[g][3];
      float pcx = dbox[p * 4 + 0], pcy = dbox[p * 4 + 1];
      float pw = dbox[p * 4 + 2], ph = dbox[p * 4 + 3];
      float t4[4];
      t4[0] = ((mx1 + mx2) * 0.5f - pcx) / (0.1f * pw);
      t4[1] = ((my1 + my2) * 0.5f - pcy) / (0.1f * ph);
      t4[2] = __logf((mx2 - mx1) / pw) * 5.0f;   // /VAR1(0.2)
      t4[3] = __logf((my2 - my1) / ph) * 5.0f;
      const float* ld = loc_data + idx * 4;
#pragma unroll
      for (int k = 0; k < 4; ++k) {
        float d = fabsf(ld[k] - t4[k]);
        ll += (d < 1.0f) ? 0.5f * d * d : (d - 0.5f);
      }
    }
  }
  float ws = warp_sum_f(ll);
  int wc = cnt;
#pragma unroll
  for (int off = 16; off; off >>= 1) wc += __shfl_xor(wc, off, 32);
  int lane = tid & 31, wid = tid >> 5;
  if (lane == 0) {
    sred[wid] = ws;
    atomicAdd(&num_pos[b], wc);  // integer atomic: deterministic
  }
  __syncthreads();
  if (tid == 0) {
    float s = 0.f;
    for (int i = 0; i < 8; ++i) s += sred[i];
    partial_l[blockIdx.y * gridDim.x + blockIdx.x] = s;
  }
}

// -------------------------------- cross entropy with WMMA f32 reduction ---
// Wave handles 16 rows. Lane = (half = lane>>4, m = lane&15). A-matrix
// 16x4 f32 layout (ISA 7.12.2): lanes 0-15 hold K0,K1; lanes 16-31 K2,K3.
// acc over c == (2*half + {0,1}) mod 4. One V_WMMA_F32_16X16X4_F32 with
// B = ones gives D[m][*] = sum_k A[m][k] = S_m in full f32.
__global__ __launch_bounds__(256) void k_ce(const float* __restrict__ conf_data,
                                            const int* __restrict__ conf_t,
                                            float* __restrict__ lfr,
                                            float* __restrict__ partial_ce) {
  __shared__ float sc[8][16 * CC];  // 8 waves * 16 rows * 81 = 41472 B
  __shared__ float sred[8];
  int tid = threadIdx.x;
  int lane = tid & 31, wid = tid >> 5;
  int hfl = lane >> 4;       // K-slot group
  int rowm = lane & 15;      // M index
  int sub = lane & 7;
  bool writer = (lane & 8) == 0;
  float posce = 0.f;
  v2f ones; ones[0] = 1.0f; ones[1] = 1.0f;

  for (int tile = blockIdx.x; tile < NTILES; tile += gridDim.x) {
    int base = tile * TILE + wid * 16;
    // Stage 16 contiguous rows = 1296 contiguous floats = 5184 bytes.
    // base is a multiple of 16 rows -> region start is 16B aligned
    // (16 * 324B = 5184B), so async B128 chunks are legal.
    const float* src = conf_data + (size_t)base * CC;
#if HAVE_ASYNC_LDS
    {
      const char* gs = (const char*)src + lane * 16;
      char* ls = (char*)&sc[wid][0] + lane * 16;
      // 324 x 16B chunks: 10 full-wave issues + 4-lane tail. The 24-bit
      // IOFFSET literal advances BOTH the global and LDS address (ISA §10
      // async ops), so address VGPRs are loop-invariant; the offset must be
      // an ICE, hence the literal expansion.
#define ALD128(OFF)                                             \
      __builtin_amdgcn_global_load_async_to_lds_b128(           \
          (gi32x4*)(void*)gs, (li32x4*)(void*)ls, (OFF), 0)
      ALD128(0);    ALD128(512);  ALD128(1024); ALD128(1536);
      ALD128(2048); ALD128(2560); ALD128(3072); ALD128(3584);
      ALD128(4096); ALD128(4608);
      if (lane < 4) ALD128(5120);
#undef ALD128
      wait_asynccnt0();
    }
#else
    for (int i = lane; i < 16 * CC; i += 32) sc[wid][i] = src[i];
#endif
    __syncthreads();

    const float* rp = &sc[wid][rowm * CC];
    int tcls = conf_t[base + rowm];
    float a0 = 0.f, a1 = 0.f, tval = 0.f;
#pragma unroll
    for (int i = 0; i < 20; ++i) {
      int c0 = 4 * i + 2 * hfl;
      int c1 = c0 + 1;
      float x0 = rp[c0];
      float x1 = rp[c1];
      a0 += __expf(x0);
      a1 += __expf(x1);
      if (c0 == tcls) tval = x0;
      if (c1 == tcls) tval = x1;
    }
    if (hfl == 0) {  // c = 80 lands in K-slot 0
      float x = rp[80];
      a0 += __expf(x);
      if (tcls == 80) tval = x;
    }
    tval += __shfl_xor(tval, 16, 32);  // full target logit on both halves

    v2f A; A[0] = a0; A[1] = a1;
    v8f Cz = {};
    v8f D = __builtin_amdgcn_wmma_f32_16x16x4_f32(
        /*neg_a=*/false, A, /*neg_b=*/false, ones,
        /*c_mod=*/(short)0, Cz, /*reuse_a=*/false, /*reuse_b=*/false);

    // D layout: VGPR r = row M=r (lanes 0-15) / M=r+8 (lanes 16-31).
    float s0 = (sub & 1) ? D[1] : D[0];
    float s1 = (sub & 1) ? D[3] : D[2];
    float s2 = (sub & 1) ? D[5] : D[4];
    float s3 = (sub & 1) ? D[7] : D[6];
    float u0 = (sub & 2) ? s1 : s0;
    float u1 = (sub & 2) ? s3 : s2;
    float S = (sub & 4) ? u1 : u0;

    int row_out = hfl * 8 + sub;
    float t_out = __shfl(tval, row_out, 32);
    int c_out = __shfl(tcls, row_out, 32);
    if (writer) {  // lanes 0-7 -> rows 0-7, lanes 16-23 -> rows 8-15
      float ce = __logf(S) - t_out;  // logsumexp - target (no max needed: |x|<~6)
      bool pos = c_out > 0;
      lfr[base + row_out] = pos ? 0.f : ce;
      posce += pos ? ce : 0.f;
    }
    __syncthreads();
  }
  float ws = warp_sum_f(posce);
  if (lane == 0) sred[wid] = ws;
  __syncthreads();
  if (tid == 0) {
    float s = 0.f;
    for (int i = 0; i < 8; ++i) s += sred[i];
    partial_ce[blockIdx.x] = s;
  }
}

// -------------------- deterministic top-K sum per row (radix select) ------
__global__ __launch_bounds__(256) void k_select(const float* __restrict__ lfr,
                                                const int* __restrict__ num_pos,
                                                float* __restrict__ topk_part) {
  __shared__ unsigned cnt[256];
  __shared__ unsigned sh_prefix, sh_remaining;
  __shared__ float sred[8];
  int b = blockIdx.x, tid = threadIdx.x;
  long long np = num_pos[b];
  unsigned K = (unsigned)((np * NEG_POS_K < (long long)PP) ? np * NEG_POS_K : PP);
  if (K == 0) {
    if (tid == 0) topk_part[b] = 0.f;
    return;
  }
  const float* row = lfr + (size_t)b * PP;
  if (tid == 0) { sh_prefix = 0; sh_remaining = K; }
  for (int pass = 0; pass < 4; ++pass) {
    int ss = 24 - 8 * pass;
    cnt[tid] = 0;
    __syncthreads();
    unsigned pref = sh_prefix;
    for (int p = tid; p < PP; p += 256) {
      unsigned u = __float_as_uint(row[p]);  // all values >= 0 -> monotonic bits
      if (pass == 0 || (u >> (ss + 8)) == pref)
        atomicAdd(&cnt[(u >> ss) & 255u], 1u);
    }
    __syncthreads();
    if (tid == 0) {
      unsigned rem = sh_remaining, above = 0;
      int T = 0;
      for (int bin = 255; bin >= 0; --bin) {
        unsigned c = cnt[bin];
        if (above + c >= rem) { T = bin; break; }
        above += c;
      }
      sh_remaining = rem - above;
      sh_prefix = (pref << 8) | (unsigned)T;
    }
    __syncthreads();
  }
  unsigned tau = sh_prefix;
  unsigned take_eq = sh_remaining;  // # of exactly-tau elements to include
  float local = 0.f;
  for (int p = tid; p < PP; p += 256) {
    float v = row[p];
    if (__float_as_uint(v) > tau) local += v;
  }
  float ws = warp_sum_f(local);
  int lane = tid & 31, wid = tid >> 5;
  if (lane == 0) sred[wid] = ws;
  __syncthreads();
  if (tid == 0) {
    float s = 0.f;
    for (int i = 0; i < 8; ++i) s += sred[i];
    topk_part[b] = s + (float)take_eq * __uint_as_float(tau);
  }
}

// ------------------------------------------------------------- finalize ---
__global__ __launch_bounds__(256) void k_final(const float* __restrict__ partial_l,
                                               const float* __restrict__ partial_ce,
                                               const float* __restrict__ topk_part,
                                               const int* __restrict__ num_pos,
                                               float* __restrict__ out,
                                               int n_pl, int n_pc) {
  __shared__ float sred[8];
  __shared__ float s_lossl;
  int tid = threadIdx.x, lane = tid & 31, wid = tid >> 5;

  float a = 0.f;
  for (int i = tid; i < n_pl; i += 256) a += partial_l[i];
  float ws = warp_sum_f(a);
  if (lane == 0) sred[wid] = ws;
  __syncthreads();
  if (tid == 0) {
    float s = 0.f;
    for (int i = 0; i < 8; ++i) s += sred[i];
    s_lossl = s;
  }
  __syncthreads();

  float c = 0.f;
  for (int i = tid; i < n_pc; i += 256) c += partial_ce[i];
  ws = warp_sum_f(c);
  if (lane == 0) sred[wid] = ws;
  __syncthreads();
  if (tid == 0) {
    float posce = 0.f;
    for (int i = 0; i < 8; ++i) posce += sred[i];
    float tk = 0.f;
    int n = 0;
    for (int b = 0; b < BB; ++b) { tk += topk_part[b]; n += num_pos[b]; }
    float N = (float)n;
    out[0] = (posce + tk) / N;  // loss_c / N
    out[1] = s_lossl / N;       // loss_l / N
  }
}

// --------------------------------------------------------------- launch ---
extern "C" void kernel_launch(void* const* d_in, const int* in_sizes, int n_in,
                              void* d_out, int out_size, void* d_ws, size_t ws_size,
                              hipStream_t stream) {
  const float* loc_data = (const float*)d_in[0];   // [B,P,4]
  const float* conf_data = (const float*)d_in[1];  // [B,P,C]
  const float* dbox = (const float*)d_in[2];       // [P,4] center form
  const float* targets = (const float*)d_in[3];    // [B,G,5]
  float* out = (float*)d_out;

  char* w = (char*)d_ws;
  size_t off = 0;
  auto alloc = [&](size_t bytes) -> void* {
    void* p = w + off;
    off = (off + bytes + 255) & ~(size_t)255;
    return p;
  };
  float* pp4 = (float*)alloc((size_t)PP * 4 * sizeof(float));
  float* bt_ov = (float*)alloc((size_t)ROWS * sizeof(float));
  int* bt_g = (int*)alloc((size_t)ROWS * sizeof(int));
  int* conf_t = (int*)alloc((size_t)ROWS * sizeof(int));
  float* lfr = (float*)alloc((size_t)ROWS * sizeof(float));
  unsigned long long* bp_key =
      (unsigned long long*)alloc((size_t)BB * GG * sizeof(unsigned long long));
  int* num_pos = (int*)alloc((size_t)BB * sizeof(int));
  float* partial_l = (float*)alloc((size_t)BB * PBLK * sizeof(float));
  float* partial_ce = (float*)alloc((size_t)CE_BLOCKS * sizeof(float));
  float* topk_part = (float*)alloc((size_t)BB * sizeof(float));

  k_init<<<PBLK, 256, 0, stream>>>(dbox, pp4, bp_key, num_pos);
  k_match<<<dim3(PBLK, BB), 256, 0, stream>>>(targets, pp4, bt_ov, bt_g, bp_key);
  k_force<<<1, 64, 0, stream>>>(bp_key, bt_ov, bt_g);
  k_conf_loc<<<dim3(PBLK, BB), 256, 0, stream>>>(targets, dbox, loc_data, bt_ov,
                                                 bt_g, conf_t, num_pos, partial_l);
  k_ce<<<CE_BLOCKS, 256, 0, stream>>>(conf_data, conf_t, lfr, partial_ce);
  k_select<<<BB, 256, 0, stream>>>(lfr, num_pos, topk_part);
  k_final<<<1, 256, 0, stream>>>(partial_l, partial_ce, topk_part, num_pos, out,
                                 BB * PBLK, CE_BLOCKS);
}